// RDF_14499809591860
// MI455X (gfx1250) — compile-verified
//
#include <hip/hip_runtime.h>
#include <math.h>

#define NBINS 400
#define NPART 512
#define NPAIR (NPART * NPART)   // 262144
#define SPLIT 8                 // pair-range chunks per bin (deterministic partials)

typedef float v2f __attribute__((ext_vector_type(2)));
typedef float v8f __attribute__((ext_vector_type(8)));

// ---------------------------------------------------------------------------
// Phase 1: all-pairs minimum-image distances -> d_ws[0 .. NPAIR)
// Mirrors the reference exactly: per-dim squared sep, fold if > box^2/4 via
// d + box^2 - 2*box*sqrt(d), then sqrt of the 3-dim sum.
// ---------------------------------------------------------------------------
__global__ void __launch_bounds__(256) rdf_dist_kernel(const float* __restrict__ pos,
                                                       float* __restrict__ dist) {
    int p = blockIdx.x * blockDim.x + threadIdx.x;
    if (p >= NPAIR) return;
    const int i = p >> 9;            // NPART = 512
    const int j = p & (NPART - 1);

    const float box     = 2.0f * 13.635f;
    const float box2    = box * box;
    const float half_sq = box2 * 0.25f;

    float acc = 0.0f;
#pragma unroll
    for (int k = 0; k < 3; ++k) {
        float t = pos[i * 3 + k] - pos[j * 3 + k];
        float d = t * t;                              // clip(d,0) is a no-op
        if (d > half_sq) d = d + box2 - 2.0f * box * sqrtf(d);
        acc += d;
    }
    dist[p] = sqrtf(acc);
}

// ---------------------------------------------------------------------------
// Phase 2: per-(bin, chunk) Gaussian-smeared partial sums.
// grid = (NBINS, SPLIT), block = 256 (8 wave32).
// Each wave feeds 64 gauss values per iteration into V_WMMA_F32_16X16X4_F32
// with A = ones(16x4): D[m,n] = sum_k B[k,n] + C[m,n], so after the loop the
// wave's grand total = (sum over all 8 C regs over all 32 lanes) / 16.
// EXEC is all-ones at every wmma (uniform loop, branchless lane math).
// ---------------------------------------------------------------------------
__global__ void __launch_bounds__(256) rdf_bin_kernel(const float* __restrict__ dist,
                                                      float* __restrict__ partial) {
    const int bin  = blockIdx.x;
    const int q    = blockIdx.y;
    const int lane = threadIdx.x & 31;
    const int wave = threadIdx.x >> 5;

    const float width = (float)(13.635 / 399.0);          // offsets[1]-offsets[0]
    const float coeff = -0.5f / (width * width);
    const float off   = (float)((double)bin * (13.635 / 399.0));

    v8f c = {0.f, 0.f, 0.f, 0.f, 0.f, 0.f, 0.f, 0.f};
    v2f a = {1.0f, 1.0f};                                 // A = ones(16x4)

    const int base  = q * (NPAIR / SPLIT);                // 32768 dists per chunk
    const int iters = (NPAIR / SPLIT) / (8 * 64);         // 64

    for (int it = 0; it < iters; ++it) {
        int idx  = base + ((it * 8 + wave) << 6) + lane;
        float d0 = dist[idx];
        float d1 = dist[idx + 32];
        float t0 = d0 - off;                              // well-conditioned in VALU
        float t1 = d1 - off;
        float g0 = (d0 > 0.0f) ? __expf(coeff * t0 * t0) : 0.0f;  // self-pair mask
        float g1 = (d1 > 0.0f) ? __expf(coeff * t1 * t1) : 0.0f;
        v2f b = {g0, g1};
        // (neg_a, A, neg_b, B, c_mod, C, reuse_a, reuse_b)
        c = __builtin_amdgcn_wmma_f32_16x16x4_f32(false, a, false, b,
                                                  (short)0, c, false, false);
    }

    float s = c[0] + c[1] + c[2] + c[3] + c[4] + c[5] + c[6] + c[7];
#pragma unroll
    for (int o = 16; o > 0; o >>= 1) s += __shfl_xor(s, o, 32);

    __shared__ float lsum[8];
    if (lane == 0) lsum[wave] = s * (1.0f / 16.0f);       // rows of D are identical
    __syncthreads();
    if (threadIdx.x == 0) {
        float tot = 0.0f;
        for (int w = 0; w < 8; ++w) tot += lsum[w];       // fixed order: deterministic
        partial[bin * SPLIT + q] = tot;
    }
}

// ---------------------------------------------------------------------------
// Phase 3: finalize. One block: sum partials per bin (fixed order), normalize,
// emit count[400], bins[401], rdf[400] into d_out (flat, tuple order).
// ---------------------------------------------------------------------------
__global__ void __launch_bounds__(512) rdf_final_kernel(const float* __restrict__ partial,
                                                        float* __restrict__ out) {
    __shared__ float cnt[NBINS];
    __shared__ float red[512];
    const int t = threadIdx.x;

    float v = 0.0f;
    if (t < NBINS) {
        float s = 0.0f;
        for (int qq = 0; qq < SPLIT; ++qq) s += partial[t * SPLIT + qq];
        cnt[t] = s;
        v = s;
    }
    red[t] = v;
    __syncthreads();
#pragma unroll
    for (int stride = 256; stride > 0; stride >>= 1) {
        if (t < stride) red[t] += red[t + stride];
        __syncthreads();
    }
    const float total = red[0];

    if (t < NBINS + 1) {
        out[NBINS + t] = (float)((double)t * (13.635 / 400.0));      // bins
    }
    if (t < NBINS) {
        float cn = cnt[t] / total;
        out[t] = cn;                                                 // count
        double b0  = (double)t       * (13.635 / 400.0);
        double b1  = (double)(t + 1) * (13.635 / 400.0);
        double Vt  = (4.0 / 3.0) * M_PI * 13.635 * 13.635 * 13.635;
        double vol = (4.0 * M_PI / 3.0) * (b1 * b1 * b1 - b0 * b0 * b0);
        out[2 * NBINS + 1 + t] = (float)((double)cn / (vol / Vt));   // rdf
    }
}

// ---------------------------------------------------------------------------
extern "C" void kernel_launch(void* const* d_in, const int* in_sizes, int n_in,
                              void* d_out, int out_size, void* d_ws, size_t ws_size,
                              hipStream_t stream) {
    const float* pos = (const float*)d_in[0];
    float* out       = (float*)d_out;
    float* dist      = (float*)d_ws;            // NPAIR floats (1 MB, L2-resident)
    float* partial   = dist + NPAIR;            // NBINS*SPLIT floats

    rdf_dist_kernel<<<NPAIR / 256, 256, 0, stream>>>(pos, dist);

    dim3 g2(NBINS, SPLIT);
    rdf_bin_kernel<<<g2, 256, 0, stream>>>(dist, partial);

    rdf_final_kernel<<<1, 512, 0, stream>>>(partial, out);
}